// GCN_3083786519229
// MI455X (gfx1250) — compile-verified
//
#include <hip/hip_runtime.h>
#include <hip/hip_bf16.h>

typedef float v2f __attribute__((ext_vector_type(2)));
typedef float v8f __attribute__((ext_vector_type(8)));

#define BM 128
#define BN 64
#define BK 16

// ---------------------------------------------------------------------------
// GEMM: C[row, col] = dinv[row] * sum_k A[row,k] * B[k,col]
// fp32 WMMA 16x16x4, block tile 128x64, 8 waves each computing 32x32.
// ---------------------------------------------------------------------------
__global__ __launch_bounds__(256) void gemm_dinv_wmma(
    const float* __restrict__ A, const float* __restrict__ B,
    float* __restrict__ C, const float* __restrict__ dinv,
    int M, int K, int Ncols)
{
    __shared__ float Alds[BM][BK + 1];   // stride 17: conflict-free A-frag reads
    __shared__ float Blds[BK][BN + 16];  // stride 80: halves hit disjoint banks

    const int tid  = threadIdx.x;
    const int wave = tid >> 5;
    const int lane = tid & 31;
    const int half = lane >> 4;
    const int lr   = lane & 15;

    const int blockM = blockIdx.x * BM;
    const int blockN = blockIdx.y * BN;
    const int waveM  = (wave >> 1) * 32;  // 0,32,64,96
    const int waveN  = (wave & 1) * 32;   // 0,32

    v8f acc[2][2];
    #pragma unroll
    for (int i = 0; i < 2; ++i)
        #pragma unroll
        for (int j = 0; j < 2; ++j)
            #pragma unroll
            for (int l = 0; l < 8; ++l) acc[i][j][l] = 0.0f;

    for (int k0 = 0; k0 < K; k0 += BK) {
        // A tile: 128x16, two float4 per thread (K is a multiple of 16)
        #pragma unroll
        for (int it = 0; it < 2; ++it) {
            int q    = tid + it * 256;   // float4 slot 0..511
            int row  = q >> 2;
            int cf4  = (q & 3) * 4;
            int grow = blockM + row;
            float4 v = make_float4(0.f, 0.f, 0.f, 0.f);
            if (grow < M) v = *(const float4*)(A + (size_t)grow * K + k0 + cf4);
            Alds[row][cf4 + 0] = v.x; Alds[row][cf4 + 1] = v.y;
            Alds[row][cf4 + 2] = v.z; Alds[row][cf4 + 3] = v.w;
        }
        // B tile: 16x64, scalar guarded loads (Ncols may be 40 < BN)
        #pragma unroll
        for (int it = 0; it < 4; ++it) {
            int p   = tid + it * 256;    // 0..1023
            int row = p >> 6;
            int col = p & 63;
            float v = 0.f;
            if (blockN + col < Ncols)
                v = B[(size_t)(k0 + row) * Ncols + blockN + col];
            Blds[row][col] = v;
        }
        __syncthreads();

        #pragma unroll
        for (int kk = 0; kk < BK; kk += 4) {
            v2f a[2], b[2];
            #pragma unroll
            for (int mt = 0; mt < 2; ++mt) {
                int rm = waveM + mt * 16 + lr;
                a[mt].x = Alds[rm][kk + half * 2 + 0];
                a[mt].y = Alds[rm][kk + half * 2 + 1];
            }
            #pragma unroll
            for (int nt = 0; nt < 2; ++nt) {
                int cn = waveN + nt * 16 + lr;
                b[nt].x = Blds[kk + half * 2 + 0][cn];
                b[nt].y = Blds[kk + half * 2 + 1][cn];
            }
            #pragma unroll
            for (int mt = 0; mt < 2; ++mt)
                #pragma unroll
                for (int nt = 0; nt < 2; ++nt)
                    acc[mt][nt] = __builtin_amdgcn_wmma_f32_16x16x4_f32(
                        false, a[mt], false, b[nt], (short)0, acc[mt][nt],
                        false, false);
        }
        __syncthreads();
    }

    // Store with per-row dinv scaling (feeds the propagation phase)
    #pragma unroll
    for (int mt = 0; mt < 2; ++mt)
        #pragma unroll
        for (int nt = 0; nt < 2; ++nt)
            #pragma unroll
            for (int i = 0; i < 8; ++i) {
                int grow = blockM + waveM + mt * 16 + i + half * 8;
                int gcol = blockN + waveN + nt * 16 + lr;
                if (grow < M && gcol < Ncols)
                    C[(size_t)grow * Ncols + gcol] = dinv[grow] * acc[mt][nt][i];
            }
}

// ---------------------------------------------------------------------------
// CSR build: counts, prefix sum, fill
// ---------------------------------------------------------------------------
__global__ void count_edges(int* cnt, const long long* __restrict__ col, int e) {
    int i = blockIdx.x * blockDim.x + threadIdx.x;
    if (i < e) atomicAdd(&cnt[(int)col[i]], 1);
}
__global__ void dinv_from_cnt(float* dinv, const int* __restrict__ cnt, int n) {
    int i = blockIdx.x * blockDim.x + threadIdx.x;
    if (i < n) dinv[i] = rsqrtf((float)cnt[i] + 1.0f);  // +1 self loop
}
// Per-block inclusive scan (Hillis-Steele); writes exclusive partials + block totals
__global__ void scan_block(const int* __restrict__ cnt, int* start, int* blockSums, int n) {
    __shared__ int sh[256];
    int i = blockIdx.x * 256 + threadIdx.x;
    int v = (i < n) ? cnt[i] : 0;
    sh[threadIdx.x] = v;
    __syncthreads();
    #pragma unroll
    for (int off = 1; off < 256; off <<= 1) {
        int t = (threadIdx.x >= (unsigned)off) ? sh[threadIdx.x - off] : 0;
        __syncthreads();
        sh[threadIdx.x] += t;
        __syncthreads();
    }
    if (i < n) start[i] = sh[threadIdx.x] - v;     // exclusive, pre-offset
    if (threadIdx.x == 255) blockSums[blockIdx.x] = sh[255];
}
__global__ void scan_offsets(int* blockSums, int nb) {
    if (blockIdx.x == 0 && threadIdx.x == 0) {
        int run = 0;
        for (int b = 0; b < nb; ++b) { int t = blockSums[b]; blockSums[b] = run; run += t; }
    }
}
__global__ void add_offsets(int* start, const int* __restrict__ blockSums,
                            int* cursor, int n, int e) {
    int i = blockIdx.x * 256 + threadIdx.x;
    if (i < n) {
        int s = start[i] + blockSums[blockIdx.x];
        start[i]  = s;
        cursor[i] = s;
    }
    if (i == 0) start[n] = e;
}
__global__ void fill_csr(int* cursor, int* esrc,
                         const long long* __restrict__ row,
                         const long long* __restrict__ col, int e) {
    int i = blockIdx.x * blockDim.x + threadIdx.x;
    if (i < e) {
        int c = (int)col[i];
        int pos = atomicAdd(&cursor[c], 1);
        esrc[pos] = (int)row[i];
    }
}

// ---------------------------------------------------------------------------
// Folded BN params: scale = g*rsqrt(v+eps); shift = scale*(b-m)+be
// ---------------------------------------------------------------------------
__global__ void bn_params(float* scale, float* shift,
                          const float* __restrict__ g, const float* __restrict__ be,
                          const float* __restrict__ m, const float* __restrict__ v,
                          const float* __restrict__ b, int h) {
    int i = blockIdx.x * blockDim.x + threadIdx.x;
    if (i < h) {
        float s  = g[i] * rsqrtf(v[i] + 1e-5f);
        scale[i] = s;
        shift[i] = s * (b[i] - m[i]) + be[i];
    }
}

// ---------------------------------------------------------------------------
// Fused gather + epilogue (deterministic, no atomics):
//   out[c] = relu(scale * (dinv[c] * (hs[c] + sum_{r in N(c)} hs[r])) + shift)
// One wave per node; each lane owns float4 chunks of the feature row.
// ---------------------------------------------------------------------------
__global__ void gather_bn_relu(float* __restrict__ out, const float* __restrict__ hs,
                               const int* __restrict__ start, const int* __restrict__ esrc,
                               const float* __restrict__ dinv,
                               const float* __restrict__ scale, const float* __restrict__ shift,
                               int n, int F) {
    int gid  = blockIdx.x * blockDim.x + threadIdx.x;
    int node = gid >> 5;
    int lane = gid & 31;
    if (node >= n) return;  // wave-uniform
    int s0 = start[node], s1 = start[node + 1];
    float d = dinv[node];
    int nchunks = F >> 2;
    for (int c0 = lane; c0 < nchunks; c0 += 32) {
        int j = c0 * 4;
        float4 acc = *(const float4*)(hs + (size_t)node * F + j);  // self loop
        for (int k = s0; k < s1; ++k) {
            int r = esrc[k];
            float4 v = *(const float4*)(hs + (size_t)r * F + j);
            acc.x += v.x; acc.y += v.y; acc.z += v.z; acc.w += v.w;
        }
        float4 sc = *(const float4*)(scale + j);
        float4 sh = *(const float4*)(shift + j);
        float4 o;
        o.x = fmaxf(sc.x * (d * acc.x) + sh.x, 0.f);
        o.y = fmaxf(sc.y * (d * acc.y) + sh.y, 0.f);
        o.z = fmaxf(sc.z * (d * acc.z) + sh.z, 0.f);
        o.w = fmaxf(sc.w * (d * acc.w) + sh.w, 0.f);
        *(float4*)(out + (size_t)node * F + j) = o;
    }
}

// Layer-3 variant: z[c] = dinv[c]*(hs[c] + sum hs[r]) + b3
__global__ void gather_bias(float* __restrict__ z, const float* __restrict__ hs,
                            const int* __restrict__ start, const int* __restrict__ esrc,
                            const float* __restrict__ dinv, const float* __restrict__ b3,
                            int n, int F) {
    int gid  = blockIdx.x * blockDim.x + threadIdx.x;
    int node = gid >> 5;
    int lane = gid & 31;
    if (node >= n) return;
    int s0 = start[node], s1 = start[node + 1];
    float d = dinv[node];
    int nchunks = F >> 2;
    for (int c0 = lane; c0 < nchunks; c0 += 32) {
        int j = c0 * 4;
        float4 acc = *(const float4*)(hs + (size_t)node * F + j);
        for (int k = s0; k < s1; ++k) {
            int r = esrc[k];
            float4 v = *(const float4*)(hs + (size_t)r * F + j);
            acc.x += v.x; acc.y += v.y; acc.z += v.z; acc.w += v.w;
        }
        float4 b = *(const float4*)(b3 + j);
        float4 o;
        o.x = d * acc.x + b.x;
        o.y = d * acc.y + b.y;
        o.z = d * acc.z + b.z;
        o.w = d * acc.w + b.w;
        *(float4*)(z + (size_t)node * F + j) = o;
    }
}

// ---------------------------------------------------------------------------
// Graph-LN stats: sum and sum of squares over all elements, f64 accumulation
// ---------------------------------------------------------------------------
__global__ void reduce_stats(const float* __restrict__ z, int total, double* stats) {
    __shared__ double s1[256];
    __shared__ double s2[256];
    double a = 0.0, b = 0.0;
    for (int i = blockIdx.x * blockDim.x + threadIdx.x; i < total;
         i += gridDim.x * blockDim.x) {
        double v = (double)z[i];
        a += v; b += v * v;
    }
    s1[threadIdx.x] = a; s2[threadIdx.x] = b;
    __syncthreads();
    for (int s = 128; s > 0; s >>= 1) {
        if ((int)threadIdx.x < s) {
            s1[threadIdx.x] += s1[threadIdx.x + s];
            s2[threadIdx.x] += s2[threadIdx.x + s];
        }
        __syncthreads();
    }
    if (threadIdx.x == 0) {
        atomicAdd(&stats[0], s1[0]);
        atomicAdd(&stats[1], s2[0]);
    }
}

// LayerNorm (graph mode) + log_softmax per row; one thread per node.
__global__ void final_ln_softmax(float* __restrict__ out, const float* __restrict__ z,
                                 const double* __restrict__ stats,
                                 const float* __restrict__ lw, const float* __restrict__ lb,
                                 int n, int C) {
    int i = blockIdx.x * blockDim.x + threadIdx.x;
    if (i >= n) return;
    double total = (double)n * (double)C;
    double mu  = stats[0] / total;
    double var = stats[1] / total - mu * mu;
    float rinv = (float)(1.0 / sqrt(var + 1e-5));
    float muf  = (float)mu;
    const float* zr = z + (size_t)i * C;
    float mx = -3.4e38f;
    for (int j = 0; j < C; ++j) {
        float t = (zr[j] - muf) * rinv * lw[j] + lb[j];
        mx = fmaxf(mx, t);
    }
    float se = 0.f;
    for (int j = 0; j < C; ++j) {
        float t = (zr[j] - muf) * rinv * lw[j] + lb[j];
        se += expf(t - mx);
    }
    float lse = logf(se);
    float* orow = out + (size_t)i * C;
    for (int j = 0; j < C; ++j) {
        float t = (zr[j] - muf) * rinv * lw[j] + lb[j];
        orow[j] = t - mx - lse;
    }
}

// ---------------------------------------------------------------------------
extern "C" void kernel_launch(void* const* d_in, const int* in_sizes, int n_in,
                              void* d_out, int out_size, void* d_ws, size_t ws_size,
                              hipStream_t stream) {
    const float*     x   = (const float*)d_in[0];
    const long long* ei  = (const long long*)d_in[1];
    const float *W1 = (const float*)d_in[2],  *b1 = (const float*)d_in[3];
    const float *g1 = (const float*)d_in[4],  *be1 = (const float*)d_in[5];
    const float *m1 = (const float*)d_in[6],  *v1 = (const float*)d_in[7];
    const float *W2 = (const float*)d_in[8],  *b2 = (const float*)d_in[9];
    const float *g2 = (const float*)d_in[10], *be2 = (const float*)d_in[11];
    const float *m2 = (const float*)d_in[12], *v2 = (const float*)d_in[13];
    const float *W3 = (const float*)d_in[14], *b3 = (const float*)d_in[15];
    const float *lw = (const float*)d_in[16], *lb = (const float*)d_in[17];

    const int HID = in_sizes[3];              // 768
    const int FIN = in_sizes[2] / HID;        // 128
    const int N   = in_sizes[0] / FIN;        // 100000
    const int E   = in_sizes[1] / 2;          // 400000
    const int NC  = in_sizes[15];             // 40

    const long long* rowIdx = ei;        // edge_index[0] = source
    const long long* colIdx = ei + E;    // edge_index[1] = target

    // Workspace carving (256B aligned)
    auto align = [](size_t v) { return (v + 255) & ~(size_t)255; };
    char* ws = (char*)d_ws;
    size_t off = 0;
    size_t bigBytes   = align((size_t)N * HID * sizeof(float));
    size_t smallBytes = align((size_t)N * NC * sizeof(float));
    const int nb = (N + 255) / 256;
    float* bufA   = (float*)(ws + off); off += bigBytes;
    float* bufB   = (float*)(ws + off); off += bigBytes;
    float* bufC   = (float*)(ws + off); off += smallBytes;
    float* bufD   = (float*)(ws + off); off += smallBytes;
    float* dinv   = (float*)(ws + off); off += align((size_t)N * sizeof(float));
    float* sc1    = (float*)(ws + off); off += align((size_t)HID * sizeof(float));
    float* sh1    = (float*)(ws + off); off += align((size_t)HID * sizeof(float));
    float* sc2    = (float*)(ws + off); off += align((size_t)HID * sizeof(float));
    float* sh2    = (float*)(ws + off); off += align((size_t)HID * sizeof(float));
    int*   cnt    = (int*)(ws + off);   off += align((size_t)N * sizeof(int));
    int*   start  = (int*)(ws + off);   off += align((size_t)(N + 1) * sizeof(int));
    int*   cursor = (int*)(ws + off);   off += align((size_t)N * sizeof(int));
    int*   esrc   = (int*)(ws + off);   off += align((size_t)E * sizeof(int));
    int*   bsums  = (int*)(ws + off);   off += align((size_t)nb * sizeof(int));
    double* stats = (double*)(ws + off); off += align(2 * sizeof(double));

    const int TB = 256;
    dim3 blk(TB);

    // 1) Degree counts -> dinv, CSR build (target -> incoming sources)
    hipMemsetAsync(cnt, 0, (size_t)N * sizeof(int), stream);
    count_edges<<<dim3((E + TB - 1) / TB), blk, 0, stream>>>(cnt, colIdx, E);
    dinv_from_cnt<<<dim3((N + TB - 1) / TB), blk, 0, stream>>>(dinv, cnt, N);
    scan_block<<<dim3(nb), blk, 0, stream>>>(cnt, start, bsums, N);
    scan_offsets<<<dim3(1), dim3(1), 0, stream>>>(bsums, nb);
    add_offsets<<<dim3(nb), blk, 0, stream>>>(start, bsums, cursor, N, E);
    fill_csr<<<dim3((E + TB - 1) / TB), blk, 0, stream>>>(cursor, esrc, rowIdx, colIdx, E);

    // 2) Folded BN params
    bn_params<<<dim3((HID + TB - 1) / TB), blk, 0, stream>>>(sc1, sh1, g1, be1, m1, v1, b1, HID);
    bn_params<<<dim3((HID + TB - 1) / TB), blk, 0, stream>>>(sc2, sh2, g2, be2, m2, v2, b2, HID);

    dim3 gemmBlk(256);
    dim3 grid1((N + BM - 1) / BM, (HID + BN - 1) / BN);
    dim3 grid3((N + BM - 1) / BM, (NC + BN - 1) / BN);
    int gatherGrid = (N * 32 + TB - 1) / TB;   // one wave per node

    // 3) Layer 1: hs1 = dinv*(x@W1) -> fused gather + BN + ReLU
    gemm_dinv_wmma<<<grid1, gemmBlk, 0, stream>>>(x, W1, bufA, dinv, N, FIN, HID);
    gather_bn_relu<<<dim3(gatherGrid), blk, 0, stream>>>(bufB, bufA, start, esrc, dinv,
                                                         sc1, sh1, N, HID);

    // 4) Layer 2
    gemm_dinv_wmma<<<grid1, gemmBlk, 0, stream>>>(bufB, W2, bufA, dinv, N, HID, HID);
    gather_bn_relu<<<dim3(gatherGrid), blk, 0, stream>>>(bufB, bufA, start, esrc, dinv,
                                                         sc2, sh2, N, HID);

    // 5) Layer 3
    gemm_dinv_wmma<<<grid3, gemmBlk, 0, stream>>>(bufB, W3, bufC, dinv, N, HID, NC);
    gather_bias<<<dim3(gatherGrid), blk, 0, stream>>>(bufD, bufC, start, esrc, dinv,
                                                      b3, N, NC);

    // 6) Graph LayerNorm + log_softmax
    hipMemsetAsync(stats, 0, 2 * sizeof(double), stream);
    int total = N * NC;
    int rgrid = (total + TB - 1) / TB; if (rgrid > 2048) rgrid = 2048;
    reduce_stats<<<dim3(rgrid), blk, 0, stream>>>(bufD, total, stats);
    final_ln_softmax<<<dim3((N + TB - 1) / TB), blk, 0, stream>>>(
        (float*)d_out, bufD, stats, lw, lb, N, NC);
}